// FUSE_74612171866758
// MI455X (gfx1250) — compile-verified
//
#include <hip/hip_runtime.h>
#include <hip/hip_bf16.h>
#include <stdint.h>

// ---------------------------------------------------------------------------
// FUSE bucket-model scan for MI455X (gfx1250, wave32).
//
// HBM-bound (~240 MB traffic -> ~10us at 23.3 TB/s). Strategy:
//   * 1 thread = 1 basin; 1 wave32 per workgroup (128 blocks -> 128 WGPs).
//   * Forcing tile [32 basins x 48 floats] DMA'd global->LDS with
//     global_load_async_to_lds_b64 th:TH_LOAD_NT (ASYNCcnt engine),
//     double-buffered, lane-contiguous addresses for coalescing.
//   * s_wait_asynccnt releases the current buffer while the next streams in.
//   * Output transposed through LDS, stored coalesced + non-temporal.
// ---------------------------------------------------------------------------

#define B_TOT   4096
#define T_TOT   3650
#define BLOCK   32                      // threads per block = exactly one wave32
#define TC      16                      // timesteps per tile
#define ROWF    (3*TC + 2)              // 50 floats: padded forcing row (bank-safe, 8B-aligned stride)
#define ROWO    (TC + 1)                // 17 floats: padded output row
#define NTILES  ((T_TOT + TC - 1)/TC)   // 229
#define F2_PER_ROW   (3*TC/2)           // 24 float2 per forcing row
#define F2_PER_TILE  (BLOCK*F2_PER_ROW) // 768
#define LOADS_PER_THREAD (F2_PER_TILE/BLOCK) // 24 async b64 issues per wave per tile
#define EPSF    1e-8f

__device__ __forceinline__ float sigmoid_fast(float x) {
    return 1.0f / (1.0f + __expf(-x));
}
__device__ __forceinline__ float bnd(float raw, float lo, float hi) {
    return lo + (hi - lo) * sigmoid_fast(raw);
}

// Async DMA: 8 bytes global -> LDS, non-temporal (forcing is streamed once).
// lds_off is a wave-relative LDS byte address (low 32 bits of the generic
// pointer to __shared__). Tracked by ASYNCcnt; completes in order with the
// wave's other async loads.
__device__ __forceinline__ void async_copy_b64_nt(uint32_t lds_off, const float* g) {
    asm volatile("global_load_async_to_lds_b64 %0, %1, off th:TH_LOAD_NT"
                 :: "v"(lds_off), "v"(g) : "memory");
}

__global__ __launch_bounds__(BLOCK) void fuse_scan_kernel(
        const float* __restrict__ init_state,   // (B,3)
        const float* __restrict__ forcing,      // (B,T,3)
        const float* __restrict__ params,       // (B,27)
        float* __restrict__ out)                // (B,T)
{
    __shared__ float sF[2][BLOCK * ROWF];   // double-buffered forcing tiles (12.8 KB)
    __shared__ float sO[BLOCK * ROWO];      // output transpose staging (2.2 KB)

    const int tid = threadIdx.x;
    const int b0  = blockIdx.x * BLOCK;
    const int b   = b0 + tid;

    // ---- per-basin bounded parameters (subset actually used) ----
    const float* pr = params + (long)b * 27;
    const float S1max  = bnd(pr[0],  50.0f,   5000.0f);
    const float S2max  = bnd(pr[1],  100.0f,  10000.0f);
    const float ku     = bnd(pr[6],  0.01f,   1000.0f);
    const float cexp   = bnd(pr[7],  1.0f,    20.0f);
    const float ks     = bnd(pr[12], 0.001f,  10000.0f);
    const float nexp   = bnd(pr[13], 1.0f,    10.0f);
    const float bexp   = bnd(pr[18], 0.001f,  3.0f);
    const float T_rain = bnd(pr[22], -2.0f,   4.0f);
    const float mrate  = bnd(pr[23], -2.0f,   4.0f);

    float S1  = init_state[b*3 + 0];
    float S2  = init_state[b*3 + 1];
    float SWE = init_state[b*3 + 2];

    // Wave-relative LDS byte offset of the forcing buffers (flat-LDS
    // addresses keep the allocation-relative offset in the low 32 bits).
    const uint32_t lds_f0   = (uint32_t)(uintptr_t)(&sF[0][0]);
    const uint32_t buf_step = (uint32_t)(BLOCK * ROWF * 4);   // 6400 B

    const long maxe = (long)B_TOT * T_TOT * 3 - 2;            // clamp for tail DMA

    // Cooperative, lane-contiguous DMA of one tile -> sF[bufsel].
    // Flat element index idx = tid + j*32 over 768 float2; (row,col)
    // maintained incrementally (idx+32 == +1 row +8 cols, carry at 24).
    auto prefetch_tile = [&](int tile, int bufsel) {
        const int t0 = tile * TC;
        int row = tid / F2_PER_ROW;
        int col = tid - row * F2_PER_ROW;
        const uint32_t base = lds_f0 + (uint32_t)bufsel * buf_step;
#pragma unroll
        for (int j = 0; j < LOADS_PER_THREAD; ++j) {
            long ge = (long)(b0 + row) * (3L * T_TOT) + (long)(3 * t0) + 2L * col;
            ge = ge > maxe ? maxe : ge;           // tail clamp (stays 8B-aligned)
            async_copy_b64_nt(base + (uint32_t)(row * (ROWF * 4) + col * 8),
                              forcing + ge);
            row += 1; col += 8;
            if (col >= F2_PER_ROW) { col -= F2_PER_ROW; row += 1; }
        }
    };

    prefetch_tile(0, 0);

    for (int tile = 0; tile < NTILES; ++tile) {
        const int t0    = tile * TC;
        const int steps = (T_TOT - t0) < TC ? (T_TOT - t0) : TC;
        const int buf   = tile & 1;

        if (tile + 1 < NTILES) {
            prefetch_tile(tile + 1, buf ^ 1);
            // 24 newest in flight belong to tile+1; <=24 => this tile's DMA done.
            asm volatile("s_wait_asynccnt 24" ::: "memory");
        } else {
            asm volatile("s_wait_asynccnt 0" ::: "memory");
        }
        __syncthreads();   // single-wave WG: S_NOP in HW, compiler fence in IR

        // ---- sequential bucket-model recurrence from LDS ----
        const float* __restrict__ Fr = &sF[buf][tid * ROWF];
#pragma unroll 4
        for (int t = 0; t < steps; ++t) {
            const float precip = Fr[3*t + 0];
            const float pet    = Fr[3*t + 1];
            const float temp   = Fr[3*t + 2];

            const float snow_frac = sigmoid_fast(T_rain - temp);
            const float snow      = precip * snow_frac;
            const float rain      = precip - snow;
            const float pot_melt  = fmaxf(0.0f, mrate * (temp - T_rain));
            const float SWE_after = SWE + snow;
            const float melt      = fminf(pot_melt, SWE_after);
            SWE = fmaxf(0.0f, SWE_after - melt);

            const float thr  = rain + melt;
            const float S1f  = fminf(S1, S1max) / S1max;
            const float Ac   = 1.0f - __powf(fmaxf(1.0f - S1f, EPSF), bexp);
            const float qsx  = Ac * thr;
            const float infl = thr - qsx;
            const float e1   = pet * S1f;
            const float q12  = ku * __powf(S1f, cexp);
            const float S2f  = S2 / S2max;
            const float qb   = ks * __powf(S2f, nexp);

            S1 = fmaxf(0.0f, S1 + (infl - e1 - q12));
            S2 = fmaxf(0.0f, S2 + (q12 - qb));

            sO[tid * ROWO + t] = qsx + qb;     // stride 17 -> conflict-free
        }
        __syncthreads();

        // ---- coalesced, non-temporal store of the output tile ----
#pragma unroll
        for (int j = 0; j < TC; ++j) {
            const int idx = tid + j * BLOCK;
            const int row = idx >> 4;          // TC == 16
            const int col = idx & (TC - 1);
            if (col < steps)
                __builtin_nontemporal_store(sO[row * ROWO + col],
                                            &out[(long)(b0 + row) * T_TOT + t0 + col]);
        }
        __syncthreads();   // sO reused next iteration
    }
}

extern "C" void kernel_launch(void* const* d_in, const int* in_sizes, int n_in,
                              void* d_out, int out_size, void* d_ws, size_t ws_size,
                              hipStream_t stream) {
    const float* init_state = (const float*)d_in[0];   // (4096, 3)
    const float* forcing    = (const float*)d_in[1];   // (4096, 3650, 3)
    const float* params     = (const float*)d_in[2];   // (4096, 27)
    float*       out        = (float*)d_out;           // (4096, 3650)

    (void)in_sizes; (void)n_in; (void)out_size; (void)d_ws; (void)ws_size;

    dim3 grid(B_TOT / BLOCK);   // 128 blocks -> spread over up to 128 WGPs
    dim3 block(BLOCK);          // 32 threads = one wave32
    fuse_scan_kernel<<<grid, block, 0, stream>>>(init_state, forcing, params, out);
}